// RMSNorm_56607668961691
// MI455X (gfx1250) — compile-verified
//
#include <hip/hip_runtime.h>

// RMSNorm over last dim (2048) of a (4, 4096, 2048) fp32 tensor.
// Bandwidth-bound: ~268 MB of HBM traffic -> ~11.5 us floor @ 23.3 TB/s.

typedef float fvec4 __attribute__((ext_vector_type(4)));

#define NCOL 2048
#define THREADS 256
#define WAVES (THREADS / 32)
#define ROWS_PER_BLOCK 4
#define EPSF 1e-5f

// wave32 butterfly sum via ds_swizzle_b32 (group-of-32 xor patterns,
// ISA ch.11: and_mask=0x1F, or_mask=0, xor_mask={16,8,4,2,1}).
__device__ __forceinline__ float wave32_sum(float v) {
    v += __int_as_float(__builtin_amdgcn_ds_swizzle(__float_as_int(v), 0x401F)); // xor 16
    v += __int_as_float(__builtin_amdgcn_ds_swizzle(__float_as_int(v), 0x201F)); // xor 8
    v += __int_as_float(__builtin_amdgcn_ds_swizzle(__float_as_int(v), 0x101F)); // xor 4
    v += __int_as_float(__builtin_amdgcn_ds_swizzle(__float_as_int(v), 0x081F)); // xor 2
    v += __int_as_float(__builtin_amdgcn_ds_swizzle(__float_as_int(v), 0x041F)); // xor 1
    return v;
}

__global__ __launch_bounds__(THREADS) void rmsnorm_f32_kernel(
    const float* __restrict__ x,
    const float* __restrict__ g,
    float* __restrict__ out,
    int n_rows)
{
    __shared__ fvec4 sg4[NCOL / 4];              // 8 KiB: staged gamma
    __shared__ fvec4 swsum[ROWS_PER_BLOCK][2];   // 8 per-wave partials per row

    const int tid  = threadIdx.x;
    const int lane = tid & 31;   // wave32
    const int wave = tid >> 5;
    const int c0   = 4 * tid;            // columns [c0 .. c0+3]
    const int c1   = c0 + (NCOL / 2);    // columns [c1 .. c1+3]

    // ---- CDNA5 async global->LDS staging of gamma (ASYNCcnt path) ----
    // The instruction offset applies to BOTH the LDS and global addresses
    // (ISA 08 §4.4), so one VGPR pair + one LDS-address VGPR covers both
    // 16B chunks of this thread's gamma columns.
    {
        unsigned lds_addr       = (unsigned)(unsigned long long)&sg4[tid];
        unsigned long long gadr = (unsigned long long)(g + c0);
        asm volatile(
            "global_load_async_to_lds_b128 %0, %1, off\n\t"
            "global_load_async_to_lds_b128 %0, %1, off offset:4096\n\t"
            "s_wait_asynccnt 0"
            :
            : "v"(lds_addr), "v"(gadr)
            : "memory");
    }
    __syncthreads();

    const fvec4 g0 = sg4[tid];
    const fvec4 g1 = sg4[tid + THREADS];

    const long long row0 = (long long)blockIdx.x * ROWS_PER_BLOCK;

    for (int r = 0; r < ROWS_PER_BLOCK; ++r) {
        const long long row = row0 + r;
        if (row >= n_rows) break;  // uniform per block: safe with barriers

        const float* __restrict__ xr = x   + row * (long long)NCOL;
        float* __restrict__ orow     = out + row * (long long)NCOL;

        // Streaming 128-bit non-temporal loads (x is read exactly once;
        // the 268 MB stream > 192 MB L2, so bypass keeps gamma resident).
        const fvec4 a = __builtin_nontemporal_load((const fvec4*)(xr + c0));
        const fvec4 b = __builtin_nontemporal_load((const fvec4*)(xr + c1));

        // Prefetch next row while we reduce (global_prefetch_b8).
        if (r + 1 < ROWS_PER_BLOCK) {
            __builtin_prefetch(xr + NCOL + c0, 0, 0);
            __builtin_prefetch(xr + NCOL + c1, 0, 0);
        }

        float s = a.x * a.x + a.y * a.y + a.z * a.z + a.w * a.w
                + b.x * b.x + b.y * b.y + b.z * b.z + b.w * b.w;

        s = wave32_sum(s);                      // 5x ds_swizzle butterfly

        if (lane == 0)
            ((float*)&swsum[r][0])[wave] = s;   // wave w -> element w of 2xfvec4
        __syncthreads();

        const fvec4 p0 = swsum[r][0];           // ds_load_b128
        const fvec4 p1 = swsum[r][1];           // ds_load_b128
        const float tot = (p0.x + p0.y) + (p0.z + p0.w)
                        + (p1.x + p1.y) + (p1.z + p1.w);

        const float inv = __builtin_amdgcn_rsqf(tot * (1.0f / NCOL) + EPSF);

        const fvec4 o0 = a * inv * g0;
        const fvec4 o1 = b * inv * g1;
        __builtin_nontemporal_store(o0, (fvec4*)(orow + c0));
        __builtin_nontemporal_store(o1, (fvec4*)(orow + c1));
        // per-row swsum slots: no trailing barrier needed (next row writes
        // distinct LDS addresses; next __syncthreads re-converges).
    }
}

extern "C" void kernel_launch(void* const* d_in, const int* in_sizes, int n_in,
                              void* d_out, int out_size, void* d_ws, size_t ws_size,
                              hipStream_t stream)
{
    const float* x = (const float*)d_in[0];
    const float* g = (const float*)d_in[1];
    float* out     = (float*)d_out;

    const int n_rows = in_sizes[0] / NCOL;                              // 16384
    const int blocks = (n_rows + ROWS_PER_BLOCK - 1) / ROWS_PER_BLOCK;  // 4096
    rmsnorm_f32_kernel<<<blocks, THREADS, 0, stream>>>(x, g, out, n_rows);
}